// DenseNew_84421877170445
// MI455X (gfx1250) — compile-verified
//
#include <hip/hip_runtime.h>
#include <hip/hip_bf16.h>

typedef __attribute__((ext_vector_type(2))) float v2f;
typedef __attribute__((ext_vector_type(8))) float v8f;

static constexpr int Bdim = 4096;   // batch rows
static constexpr int Ddim = 4096;   // inner K
static constexpr int Udim = 4096;   // output cols
static constexpr float QF = 2291.0f;
static constexpr double QD = 2291.0;

static constexpr int BM = 64;        // block tile M
static constexpr int BN = 128;       // block tile N
static constexpr int KC = 16;        // K chunk staged in LDS (double buffered)
static constexpr int LDA = KC + 4;   // 20 floats -> 80B row stride (16B aligned)
static constexpr int LDW = BN + 4;   // 132 floats -> 528B row stride (16B aligned)
static constexpr int NCHUNK = Ddim / KC;

// -------------------------------------------------------------------------
// CDNA5 async global->LDS copy (ASYNCcnt tracked, no VGPR data movement).
// VDST = per-lane 32-bit LDS byte address, VADDR = 64-bit global address.
// Each instruction moves 32 lanes * 16B = 512B.
// -------------------------------------------------------------------------
__device__ __forceinline__ void async_b128(unsigned lds_addr, const float* gaddr) {
    asm volatile("global_load_async_to_lds_b128 %0, %1, off"
                 :: "v"(lds_addr), "v"(gaddr) : "memory");
}

// ---------------------------------------------------------------------------
// Elementwise digit decomposition: xb = x + r; low = floormod(xb, Q);
// high = rint((xb - low)/Q)   (jnp.mod sign-of-divisor + RNE round semantics)
// ---------------------------------------------------------------------------
__global__ void prep_kernel(const float* __restrict__ x, const float* __restrict__ r,
                            float* __restrict__ low, float* __restrict__ high, int n) {
    int i = blockIdx.x * blockDim.x + threadIdx.x;
    if (i >= n) return;
    float xb = x[i] + r[i];
    float lo = fmodf(xb, QF);
    lo = (lo < 0.0f) ? (lo + QF) : lo;
    float hi = rintf((xb - lo) / QF);
    low[i]  = lo;
    high[i] = hi;
}

// ---------------------------------------------------------------------------
// Fused triple GEMM (low@W, high@W, r@W share one pass over W), double
// buffered via async global->LDS loads, f64 combine in the epilogue.
// 8 wave32s per block (2x4), each wave owns a 32x32 tile = 2x2 WMMA subtiles.
// ---------------------------------------------------------------------------
__global__ __launch_bounds__(256)
void fused_gemm_kernel(const float* __restrict__ Alow, const float* __restrict__ Ahigh,
                       const float* __restrict__ Ar,   const float* __restrict__ Wg,
                       const float* __restrict__ bias, float* __restrict__ out) {
    __shared__ float sA[2][3][BM][LDA];   // 30720 B
    __shared__ float sW[2][KC][LDW];      // 16896 B

    const int tid  = threadIdx.x;
    const int lane = tid & 31;
    const int wave = tid >> 5;
    const int wm   = wave >> 2;   // 0..1 -> M offset wm*32
    const int wn   = wave & 3;    // 0..3 -> N offset wn*32

    const int m0 = blockIdx.y * BM;
    const int n0 = blockIdx.x * BN;

    v8f accL[2][2] = {};
    v8f accH[2][2] = {};
    v8f accR[2][2] = {};

    const float* gA[3] = { Alow, Ahigh, Ar };

    // Per-chunk staging: exactly 5 async b128 wave-copies per wave.
    //  - A panels: mat t in {0,1,2}, rows wave*8 .. wave*8+7 (8 rows x 16 cols = 512B)
    //  - W panel : rows `wave` and `wave+8` (128 floats = 512B each)
    auto stage = [&](int kc, int buf) {
        const int arow = wave * 8 + (lane >> 2);
        const int ac4  = lane & 3;
        #pragma unroll
        for (int t = 0; t < 3; ++t) {
            async_b128((unsigned)(uintptr_t)&sA[buf][t][arow][ac4 * 4],
                       gA[t] + (size_t)(m0 + arow) * Ddim + kc + ac4 * 4);
        }
        #pragma unroll
        for (int t = 0; t < 2; ++t) {
            const int s = wave + t * 8;
            async_b128((unsigned)(uintptr_t)&sW[buf][s][lane * 4],
                       Wg + (size_t)(kc + s) * Udim + n0 + lane * 4);
        }
    };

    stage(0, 0);   // prologue: chunk 0 -> buffer 0

    for (int i = 0; i < NCHUNK; ++i) {
        const int buf = i & 1;
        __syncthreads();   // all waves done computing from buf^1 (chunk i-1)

        if (i + 1 < NCHUNK) {
            stage((i + 1) * KC, buf ^ 1);                        // prefetch next chunk
            asm volatile("s_wait_asynccnt 0x5" ::: "memory");    // chunk i landed; next 5 in flight
        } else {
            asm volatile("s_wait_asynccnt 0x0" ::: "memory");    // drain final chunk
        }
        __syncthreads();   // chunk i fully staged for every wave

        // --- compute: 4 k-steps of 4, 12 WMMAs per step ---
        #pragma unroll
        for (int kk = 0; kk < KC; kk += 4) {
            const int krow = kk + ((lane >= 16) ? 2 : 0);   // 32-bit 16x4 A/B frag layout
            const int sl   = lane & 15;

            v2f aL[2], aH[2], aR[2];
            #pragma unroll
            for (int ii = 0; ii < 2; ++ii) {
                const int m = wm * 32 + ii * 16 + sl;
                aL[ii] = *(const v2f*)&sA[buf][0][m][krow];
                aH[ii] = *(const v2f*)&sA[buf][1][m][krow];
                aR[ii] = *(const v2f*)&sA[buf][2][m][krow];
            }
            v2f bf[2];
            #pragma unroll
            for (int j = 0; j < 2; ++j) {
                const int n = wn * 32 + j * 16 + sl;
                bf[j].x = sW[buf][krow][n];
                bf[j].y = sW[buf][krow + 1][n];
            }
            #pragma unroll
            for (int ii = 0; ii < 2; ++ii) {
                #pragma unroll
                for (int j = 0; j < 2; ++j) {
                    accL[ii][j] = __builtin_amdgcn_wmma_f32_16x16x4_f32(
                        false, aL[ii], false, bf[j], (short)0, accL[ii][j], false, false);
                    accH[ii][j] = __builtin_amdgcn_wmma_f32_16x16x4_f32(
                        false, aH[ii], false, bf[j], (short)0, accH[ii][j], false, false);
                    accR[ii][j] = __builtin_amdgcn_wmma_f32_16x16x4_f32(
                        false, aR[ii], false, bf[j], (short)0, accR[ii][j], false, false);
                }
            }
        }
    }

    // --- epilogue: (f64)(low@W + b) + Q*(f64)(high@W) -> f32; (- r@W) / 256 ---
    #pragma unroll
    for (int ii = 0; ii < 2; ++ii) {
        #pragma unroll
        for (int j = 0; j < 2; ++j) {
            const int n     = n0 + wn * 32 + j * 16 + (lane & 15);
            const float bn  = bias[n];
            const int mbase = m0 + wm * 32 + ii * 16 + ((lane >= 16) ? 8 : 0);
            #pragma unroll
            for (int v = 0; v < 8; ++v) {
                float ol   = accL[ii][j][v] + bn;           // bias_add on low digits
                float oh   = accH[ii][j][v];
                float comb = (float)((double)ol + QD * (double)oh);
                float o    = (comb - accR[ii][j][v]) * (1.0f / 256.0f);
                out[(size_t)(mbase + v) * Udim + n] = o;
            }
        }
    }
}

// ---------------------------------------------------------------------------
extern "C" void kernel_launch(void* const* d_in, const int* in_sizes, int n_in,
                              void* d_out, int out_size, void* d_ws, size_t ws_size,
                              hipStream_t stream) {
    const float* x  = (const float*)d_in[0];   // [B,D]
    const float* r  = (const float*)d_in[1];   // [B,D]
    const float* Wg = (const float*)d_in[2];   // [D,U]
    const float* b  = (const float*)d_in[3];   // [U]
    float* out = (float*)d_out;

    float* low  = (float*)d_ws;                   // [B,D]
    float* high = low + (size_t)Bdim * Ddim;      // [B,D]

    const int n = Bdim * Ddim;
    prep_kernel<<<(n + 255) / 256, 256, 0, stream>>>(x, r, low, high, n);

    dim3 grid(Udim / BN, Bdim / BM);   // (32, 64)
    fused_gemm_kernel<<<grid, 256, 0, stream>>>(low, high, r, Wg, b, out);
}